// EdgeLearnGNN_37993280701070
// MI455X (gfx1250) — compile-verified
//
#include <hip/hip_runtime.h>
#include <math.h>

// Problem constants (match the reference)
#define B_  16
#define N_  64
#define F_  1024
#define K_  13
#define C_  2
#define EPS_BN 1e-5f

typedef __attribute__((ext_vector_type(2))) float v2f;
typedef __attribute__((ext_vector_type(8))) float v8f;

// Workspace layout (float offsets)
#define WS_S      0                         // s[b,i,j]            : 65536
#define WS_AI     (WS_S  + B_*N_*N_)        // adj + I             : 65536
#define WS_H1     (WS_AI + B_*N_*N_)        // h1[b,i,f]           : 1048576
#define WS_H2     (WS_H1 + B_*N_*F_)        // h2[b,n,f]           : 1048576
#define WS_M1     (WS_H2 + B_*N_*F_)        // bn1 mean[64]
#define WS_V1     (WS_M1 + N_)              // bn1 var[64]
#define WS_Y      (WS_V1 + N_)              // y[b,f]              : 16384
#define WS_YC     (WS_Y  + B_*F_)           // conv out            : 16384
#define WS_M2     (WS_YC + B_*F_)           // bn2 mean, var       : 2

__device__ inline v8f wmma_f32_16x16x4(v2f a, v2f b, v8f c) {
    // D = A(16x4,f32) x B(4x16,f32) + C(16x16,f32)
    return __builtin_amdgcn_wmma_f32_16x16x4_f32(
        /*neg_a=*/false, a, /*neg_b=*/false, b,
        /*c_mod=*/(short)0, c, /*reuse_a=*/false, /*reuse_b=*/false);
}

// ---------------------------------------------------------------------------
// K1: s[b,i,j] = -relu( sum_f |x[b,i,f]-x[b,j,f]| * w[f] )
// grid = B*N (one block per (b,i)), block = 256
// ---------------------------------------------------------------------------
__global__ __launch_bounds__(256)
void k1_pair_score(const float* __restrict__ x, const float* __restrict__ w,
                   float* __restrict__ s_ws) {
    __shared__ float sXi[F_];
    __shared__ float sW[F_];
    __shared__ float red[256];
    const int bid = blockIdx.x;
    const int b = bid >> 6;          // /64
    const int i = bid & 63;
    const int tid = threadIdx.x;
    const float* xb = x + (size_t)b * N_ * F_;

    for (int f = tid; f < F_; f += 256) {
        sXi[f] = xb[i * F_ + f];
        sW[f]  = w[f];
    }
    __syncthreads();

    const int j = tid & 63;
    const int c = tid >> 6;          // 0..3, each handles 256 f values
    const float* xj = xb + j * F_;
    float acc = 0.0f;
    const int f0 = c * 256;
    #pragma unroll 4
    for (int f = f0; f < f0 + 256; ++f) {
        acc += fabsf(sXi[f] - xj[f]) * sW[f];
    }
    red[tid] = acc;
    __syncthreads();
    if (tid < 64) {
        float s = red[tid] + red[tid + 64] + red[tid + 128] + red[tid + 192];
        s = -fmaxf(s, 0.0f);
        s_ws[(b * N_ + i) * N_ + tid] = s;   // tid == j
    }
}

// ---------------------------------------------------------------------------
// K2: adj[b,i,j] = softmax over i of s[b,i,j]; written to d_out+32.
// Also writes adjI = adj + I into workspace so the WMMA kernel's A-side is
// a pure load stream (no cndmask/identity VALU in the inner loop).
// grid = 4, block = 256 -> one thread per (b,j)
// ---------------------------------------------------------------------------
__global__ __launch_bounds__(256)
void k2_softmax_dim1(const float* __restrict__ s_ws, float* __restrict__ adj,
                     float* __restrict__ adjI) {
    const int t = blockIdx.x * 256 + threadIdx.x;     // 0..1023
    const int b = t >> 6;
    const int j = t & 63;
    const float* sb = s_ws + b * N_ * N_ + j;
    float mx = -1e30f;
    for (int i = 0; i < N_; ++i) mx = fmaxf(mx, sb[i * N_]);
    float sum = 0.0f;
    for (int i = 0; i < N_; ++i) sum += __expf(sb[i * N_] - mx);
    const float inv = 1.0f / sum;
    float* ab  = adj  + b * N_ * N_ + j;
    float* aib = adjI + b * N_ * N_ + j;
    for (int i = 0; i < N_; ++i) {
        const float v = __expf(sb[i * N_] - mx) * inv;
        ab[i * N_]  = v;
        aib[i * N_] = v + ((i == j) ? 1.0f : 0.0f);
    }
}

// ---------------------------------------------------------------------------
// K3: h1[b,i,f] = sum_j adjI[b,i,j] * x[b,j,f]     (WMMA f32 16x16x4)
// Per batch: M=64(i), K=64(j), N=1024(f). 4096 wave-tiles, 8 waves/block.
// The 8 waves of a block share one A tile (same b,mt) -> staged in LDS with
// a row stride of 68 floats (4*row + k bank mapping: conflict-free b64 reads).
// grid = 512, block = 256
// ---------------------------------------------------------------------------
#define SA_STRIDE 68
__global__ __launch_bounds__(256)
void k3_gin_aggregate(const float* __restrict__ adjI, const float* __restrict__ x,
                      float* __restrict__ h1) {
    __shared__ float sA[16 * SA_STRIDE];
    const int tid  = threadIdx.x;
    const int lane = tid & 31;
    const int tile0 = blockIdx.x * 8;               // 8 consecutive ft share (b,mt)
    const int b  = tile0 >> 8;
    const int mt = (tile0 & 255) >> 6;              // 0..3  (i tile)
    const int ft = (tile0 & 63) + (tid >> 5);       // per-wave f tile

    // stage A tile: rows i = mt*16 .. mt*16+15, all 64 k (j) values
    {
        const float* asrc = adjI + (size_t)b * N_ * N_ + (size_t)mt * 16 * N_;
        for (int idx = tid; idx < 16 * N_; idx += 256) {
            const int r = idx >> 6;
            const int k = idx & 63;
            sA[r * SA_STRIDE + k] = asrc[r * N_ + k];
        }
    }
    __syncthreads();

    const int row = lane & 15;
    const int hi  = lane >> 4;                      // 0: K=k0..k0+1, 1: K=k0+2..k0+3
    const float* xb = x + (size_t)b * N_ * F_;

    v8f acc;
    #pragma unroll
    for (int q = 0; q < 8; ++q) acc[q] = 0.0f;

    const int fc = ft * 16 + row;                   // B col (f)
    #pragma unroll 8
    for (int k0 = 0; k0 < N_; k0 += 4) {
        const int ka = k0 + hi * 2;
        const v2f a = *(const v2f*)(&sA[row * SA_STRIDE + ka]);
        v2f bb;
        bb.x = xb[(size_t)ka * F_ + fc];
        bb.y = xb[(size_t)(ka + 1) * F_ + fc];
        acc = wmma_f32_16x16x4(a, bb, acc);
    }

    const int col   = lane & 15;
    const int mbase = mt * 16 + hi * 8;
    float* out = h1 + ((size_t)b * N_ + mbase) * F_ + ft * 16 + col;
    #pragma unroll
    for (int q = 0; q < 8; ++q) out[(size_t)q * F_] = acc[q];
}

// ---------------------------------------------------------------------------
// K4: h2[m,o] = sum_f h1[m,f] * gin_W[o,f] + gin_b[o]   (WMMA f32 16x16x4)
// M = B*N = 1024, K = 1024, N(out) = 1024. 4096 tiles, 256 WMMAs/tile.
// Block = 8 waves handling 8 M-tiles that share one N-tile -> the 16 gin_W
// rows (64 KB) are staged once per block into LDS (stride 1028 floats so a
// B-fragment's 32 b64 lane reads hit 64 distinct banks). A-side is aligned
// v2f global loads with a one-k-block-ahead prefetch.
// grid = 512, block = 256
// ---------------------------------------------------------------------------
#define SB_STRIDE 1028
__global__ __launch_bounds__(256)
void k4_gin_linear(const float* __restrict__ h1, const float* __restrict__ ginW,
                   const float* __restrict__ ginb, float* __restrict__ h2) {
    __shared__ float sB[16 * SB_STRIDE];            // ~64.3 KB of the 320 KB WGP LDS
    const int tid  = threadIdx.x;
    const int lane = tid & 31;
    const int nt = blockIdx.x & 63;                 // shared N-tile
    const int mt = (blockIdx.x >> 6) * 8 + (tid >> 5);

    // stage B: gin_W rows nt*16 .. nt*16+15, full K
    {
        const float* bsrc = ginW + (size_t)nt * 16 * F_;
        for (int idx = tid; idx < 16 * F_; idx += 256) {
            const int r = idx >> 10;
            const int k = idx & (F_ - 1);
            sB[r * SB_STRIDE + k] = bsrc[(size_t)r * F_ + k];
        }
    }
    __syncthreads();

    const int row = lane & 15;
    const int hi  = lane >> 4;

    v8f acc;
    #pragma unroll
    for (int q = 0; q < 8; ++q) acc[q] = 0.0f;

    const float* arow = h1 + (size_t)(mt * 16 + row) * F_;   // A: h1[m, k]
    const float* brow = &sB[row * SB_STRIDE];                // B: LDS-staged gin_W[n, k]

    for (int kb = 0; kb < F_; kb += 64) {
        __builtin_prefetch(arow + kb + 64, 0, 1);            // global_prefetch_b8
        #pragma unroll
        for (int ko = 0; ko < 64; ko += 4) {
            const int ka = kb + ko + hi * 2;
            const v2f a  = *(const v2f*)(arow + ka);         // global_load_b64
            const v2f bb = *(const v2f*)(brow + ka);         // ds_load_b64
            acc = wmma_f32_16x16x4(a, bb, acc);
        }
    }

    const int col = lane & 15;
    const int n   = nt * 16 + col;
    const float bias = ginb[n];
    const int mbase = mt * 16 + hi * 8;
    float* out = h2 + (size_t)mbase * F_ + n;
    #pragma unroll
    for (int q = 0; q < 8; ++q) out[(size_t)q * F_] = acc[q] + bias;
}

// ---------------------------------------------------------------------------
// K5: BN1 batch stats per channel n over (b,f): mean, biased var
// grid = 64 (channel), block = 256
// ---------------------------------------------------------------------------
__global__ __launch_bounds__(256)
void k5_bn1_stats(const float* __restrict__ h2, float* __restrict__ mean,
                  float* __restrict__ var) {
    __shared__ float rs[256], rq[256];
    const int n = blockIdx.x;
    const int tid = threadIdx.x;
    float s = 0.0f, q = 0.0f;
    for (int idx = tid; idx < B_ * F_; idx += 256) {
        const int b = idx >> 10;
        const int f = idx & (F_ - 1);
        const float v = h2[((size_t)b * N_ + n) * F_ + f];
        s += v; q += v * v;
    }
    rs[tid] = s; rq[tid] = q;
    __syncthreads();
    for (int off = 128; off > 0; off >>= 1) {
        if (tid < off) { rs[tid] += rs[tid + off]; rq[tid] += rq[tid + off]; }
        __syncthreads();
    }
    if (tid == 0) {
        const float m = rs[0] / (float)(B_ * F_);
        mean[n] = m;
        var[n]  = rq[0] / (float)(B_ * F_) - m * m;
    }
}

// ---------------------------------------------------------------------------
// K6: y[b,f] = sum_n bn1(h2[b,n,f]) * lc_W[n] + lc_b   (BN folded into cA/cB)
// grid = 64, block = 256 -> one thread per (b,f)
// ---------------------------------------------------------------------------
__global__ __launch_bounds__(256)
void k6_channel_mix(const float* __restrict__ h2, const float* __restrict__ mean,
                    const float* __restrict__ var, const float* __restrict__ g,
                    const float* __restrict__ beta, const float* __restrict__ lcW,
                    const float* __restrict__ lcb, float* __restrict__ y) {
    __shared__ float cA[N_], cB[N_];
    const int tid = threadIdx.x;
    if (tid < N_) {
        const float inv = 1.0f / sqrtf(var[tid] + EPS_BN);
        const float gi  = g[tid] * inv;
        cA[tid] = gi * lcW[tid];
        cB[tid] = (beta[tid] - mean[tid] * gi) * lcW[tid];
    }
    __syncthreads();
    const int t = blockIdx.x * 256 + tid;    // 0..16383
    const int b = t >> 10;
    const int f = t & (F_ - 1);
    float acc = lcb[0];
    const float* hb = h2 + (size_t)b * N_ * F_ + f;
    #pragma unroll 8
    for (int n = 0; n < N_; ++n) acc += hb[(size_t)n * F_] * cA[n] + cB[n];
    y[b * F_ + f] = acc;
}

// ---------------------------------------------------------------------------
// K7: 1D conv, kernel 13, SAME (pad 6), + bias
// ---------------------------------------------------------------------------
__global__ __launch_bounds__(256)
void k7_conv(const float* __restrict__ y, const float* __restrict__ cw,
             const float* __restrict__ cb, float* __restrict__ yc) {
    const int t = blockIdx.x * 256 + threadIdx.x;
    const int b = t >> 10;
    const int f = t & (F_ - 1);
    float acc = cb[0];
    #pragma unroll
    for (int k = 0; k < K_; ++k) {
        const int src = f + k - 6;
        if (src >= 0 && src < F_) acc += y[b * F_ + src] * cw[k];
    }
    yc[b * F_ + f] = acc;
}

// ---------------------------------------------------------------------------
// K8: BN2 stats over all (b,f): single channel
// ---------------------------------------------------------------------------
__global__ __launch_bounds__(256)
void k8_bn2_stats(const float* __restrict__ yc, float* __restrict__ st) {
    __shared__ float rs[256], rq[256];
    const int tid = threadIdx.x;
    float s = 0.0f, q = 0.0f;
    for (int idx = tid; idx < B_ * F_; idx += 256) {
        const float v = yc[idx];
        s += v; q += v * v;
    }
    rs[tid] = s; rq[tid] = q;
    __syncthreads();
    for (int off = 128; off > 0; off >>= 1) {
        if (tid < off) { rs[tid] += rs[tid + off]; rq[tid] += rq[tid + off]; }
        __syncthreads();
    }
    if (tid == 0) {
        const float m = rs[0] / (float)(B_ * F_);
        st[0] = m;
        st[1] = rq[0] / (float)(B_ * F_) - m * m;
    }
}

// ---------------------------------------------------------------------------
// K9: head: bn2 -> relu -> logits[b,c] -> softmax -> d_out[0:32]
// grid = 16 (batch), block = 256
// ---------------------------------------------------------------------------
__global__ __launch_bounds__(256)
void k9_head(const float* __restrict__ yc, const float* __restrict__ st,
             const float* __restrict__ g2, const float* __restrict__ b2,
             const float* __restrict__ outW, const float* __restrict__ outb,
             float* __restrict__ probs) {
    __shared__ float r0[256], r1[256];
    const int b = blockIdx.x;
    const int tid = threadIdx.x;
    const float m2  = st[0];
    const float inv = 1.0f / sqrtf(st[1] + EPS_BN);
    const float gg  = g2[0] * inv;
    const float bb  = b2[0];
    float a0 = 0.0f, a1 = 0.0f;
    for (int f = tid; f < F_; f += 256) {
        float v = (yc[b * F_ + f] - m2) * gg + bb;
        v = fmaxf(v, 0.0f);
        a0 += v * outW[f];
        a1 += v * outW[F_ + f];
    }
    r0[tid] = a0; r1[tid] = a1;
    __syncthreads();
    for (int off = 128; off > 0; off >>= 1) {
        if (tid < off) { r0[tid] += r0[tid + off]; r1[tid] += r1[tid + off]; }
        __syncthreads();
    }
    if (tid == 0) {
        const float l0 = r0[0] + outb[0];
        const float l1 = r1[0] + outb[1];
        const float mx = fmaxf(l0, l1);
        const float e0 = __expf(l0 - mx);
        const float e1 = __expf(l1 - mx);
        const float s  = e0 + e1;
        probs[b * 2 + 0] = e0 / s;
        probs[b * 2 + 1] = e1 / s;
    }
}

// ---------------------------------------------------------------------------
extern "C" void kernel_launch(void* const* d_in, const int* in_sizes, int n_in,
                              void* d_out, int out_size, void* d_ws, size_t ws_size,
                              hipStream_t stream) {
    const float* x     = (const float*)d_in[0];
    const float* w     = (const float*)d_in[1];
    const float* ginW  = (const float*)d_in[2];
    const float* ginb  = (const float*)d_in[3];
    const float* bn1g  = (const float*)d_in[4];
    const float* bn1b  = (const float*)d_in[5];
    const float* lcW   = (const float*)d_in[6];
    const float* lcb   = (const float*)d_in[7];
    const float* convW = (const float*)d_in[8];
    const float* convb = (const float*)d_in[9];
    const float* bn2g  = (const float*)d_in[10];
    const float* bn2b  = (const float*)d_in[11];
    const float* outW  = (const float*)d_in[12];
    const float* outb  = (const float*)d_in[13];

    float* out   = (float*)d_out;
    float* probs = out;             // [16,2]
    float* adj   = out + B_ * C_;   // [16,64,64]

    float* ws   = (float*)d_ws;
    float* sws  = ws + WS_S;
    float* adjI = ws + WS_AI;
    float* h1   = ws + WS_H1;
    float* h2   = ws + WS_H2;
    float* m1   = ws + WS_M1;
    float* v1   = ws + WS_V1;
    float* y    = ws + WS_Y;
    float* yc   = ws + WS_YC;
    float* st2  = ws + WS_M2;

    k1_pair_score   <<<B_ * N_, 256, 0, stream>>>(x, w, sws);
    k2_softmax_dim1 <<<4, 256, 0, stream>>>(sws, adj, adjI);
    k3_gin_aggregate<<<512, 256, 0, stream>>>(adjI, x, h1);
    k4_gin_linear   <<<512, 256, 0, stream>>>(h1, ginW, ginb, h2);
    k5_bn1_stats    <<<N_, 256, 0, stream>>>(h2, m1, v1);
    k6_channel_mix  <<<64, 256, 0, stream>>>(h2, m1, v1, bn1g, bn1b, lcW, lcb, y);
    k7_conv         <<<64, 256, 0, stream>>>(y, convW, convb, yc);
    k8_bn2_stats    <<<1, 256, 0, stream>>>(yc, st2);
    k9_head         <<<B_, 256, 0, stream>>>(yc, st2, bn2g, bn2b, outW, outb, probs);
}